// SimpleGNN_18743237280053
// MI455X (gfx1250) — compile-verified
//
#include <hip/hip_runtime.h>
#include <hip/hip_bf16.h>
#include <math.h>

// ---------------------------------------------------------------------------
// SimpleGNN (2-layer GCN + mean pool + sigmoid head) for MI455X / gfx1250.
// - Dense transforms: V_WMMA_F32_16X16X4_F32 (wave32, one 16-node tile/wave).
// - Edge aggregation: GLOBAL_ATOMIC_ADD_F32 into accumulators that fit in
//   the 192MB L2 (agg1=16MB, agg2=32MB).
// - Streamed-once data (edge index, x, batch) loaded non-temporal so the
//   64MB index stream does not evict the hot L2-resident accumulators.
// ---------------------------------------------------------------------------

typedef float v2f __attribute__((ext_vector_type(2)));
typedef float v8f __attribute__((ext_vector_type(8)));

// ---------------- degree: deg[dst] += 1 ------------------------------------
__global__ void deg_kernel(const int* __restrict__ ei, float* __restrict__ deg, int E) {
    int e = blockIdx.x * blockDim.x + threadIdx.x;
    if (e >= E) return;
    int d = __builtin_nontemporal_load(ei + E + e);   // edge_index[1][e], NT
    atomicAdd(&deg[d], 1.0f);
}

// ---------------- dinv = rsqrt(deg + 1) ------------------------------------
__global__ void dinv_kernel(const float* __restrict__ deg, float* __restrict__ dinv, int N) {
    int i = blockIdx.x * blockDim.x + threadIdx.x;
    if (i >= N) return;
    dinv[i] = __frsqrt_rn(deg[i] + 1.0f);
}

// ---------------- layer 1 GEMM: h1 = x[N,4] @ W1[4,16] via WMMA ------------
// One wave per 16-row tile. A-layout (32-bit 16x4): lane<16 -> K0/K1,
// lane>=16 -> K2/K3, row = lane&15. B mirrored. D: VGPR r -> M=r(+8), N=lane&15.
__global__ void l1_wmma_kernel(const float* __restrict__ x, const float* __restrict__ W1,
                               float* __restrict__ h1, int ntiles) {
    int wave = (blockIdx.x * blockDim.x + threadIdx.x) >> 5;
    int lane = threadIdx.x & 31;
    if (wave >= ntiles) return;
    int row   = lane & 15;
    int khalf = (lane >> 4) << 1;          // 0 or 2
    long base = (long)wave * 16;

    v2f a;
    const float* xr = x + (base + row) * 4 + khalf;
    a.x = __builtin_nontemporal_load(xr + 0);   // x read exactly once: NT
    a.y = __builtin_nontemporal_load(xr + 1);

    v2f b;
    b.x = W1[khalf * 16 + row];            // row here doubles as N-column index
    b.y = W1[(khalf + 1) * 16 + row];

    v8f c = {};
    c = __builtin_amdgcn_wmma_f32_16x16x4_f32(false, a, false, b, (short)0, c, false, false);

    int mhi = (lane >> 4) * 8;
    #pragma unroll
    for (int r = 0; r < 8; ++r) {
        h1[(base + r + mhi) * 16 + row] = c[r];
    }
}

// ---------------- layer 2 GEMM: h2 = hin[N,16] @ W2[16,32] via WMMA --------
// Per wave: 4 chained K-steps x 2 N-tiles = 8 WMMAs.
__global__ void l2_wmma_kernel(const float* __restrict__ hin, const float* __restrict__ W2,
                               float* __restrict__ h2, int ntiles) {
    int wave = (blockIdx.x * blockDim.x + threadIdx.x) >> 5;
    int lane = threadIdx.x & 31;
    if (wave >= ntiles) return;
    int col   = lane & 15;                 // serves as both A-row and B/D-column
    int khalf = (lane >> 4) << 1;
    long base = (long)wave * 16;

    v8f acc0 = {};
    v8f acc1 = {};
    #pragma unroll
    for (int k0 = 0; k0 < 16; k0 += 4) {
        v2f a;
        const float* ar = hin + (base + col) * 16 + k0 + khalf;
        a.x = ar[0];
        a.y = ar[1];

        v2f b0, b1;
        b0.x = W2[(k0 + khalf) * 32 + col];
        b0.y = W2[(k0 + khalf + 1) * 32 + col];
        b1.x = W2[(k0 + khalf) * 32 + 16 + col];
        b1.y = W2[(k0 + khalf + 1) * 32 + 16 + col];

        acc0 = __builtin_amdgcn_wmma_f32_16x16x4_f32(false, a, false, b0, (short)0, acc0, false, false);
        acc1 = __builtin_amdgcn_wmma_f32_16x16x4_f32(false, a, false, b1, (short)0, acc1, false, false);
    }

    int mhi = (lane >> 4) * 8;
    #pragma unroll
    for (int r = 0; r < 8; ++r) {
        long nrow = base + r + mhi;
        h2[nrow * 32 + col]      = acc0[r];
        h2[nrow * 32 + 16 + col] = acc1[r];
    }
}

// ---------------- edge scatter, 16 features (4 threads/edge, float4 each) --
__global__ void edge16_kernel(const int* __restrict__ ei, const float* __restrict__ dinv,
                              const float* __restrict__ h, float* __restrict__ agg, int E) {
    long t = (long)blockIdx.x * blockDim.x + threadIdx.x;
    int e = (int)(t >> 2);
    int g = (int)(t & 3);
    if (e >= E) return;
    int s = __builtin_nontemporal_load(ei + e);        // streamed once: NT
    int d = __builtin_nontemporal_load(ei + E + e);    // streamed once: NT
    float nrm = dinv[s] * dinv[d];
    const float4 v = *((const float4*)(h + (long)s * 16) + g);
    float* ap = agg + (long)d * 16 + g * 4;
    atomicAdd(ap + 0, v.x * nrm);
    atomicAdd(ap + 1, v.y * nrm);
    atomicAdd(ap + 2, v.z * nrm);
    atomicAdd(ap + 3, v.w * nrm);
}

// ---------------- edge scatter, 32 features (8 threads/edge) ---------------
__global__ void edge32_kernel(const int* __restrict__ ei, const float* __restrict__ dinv,
                              const float* __restrict__ h, float* __restrict__ agg, int E) {
    long t = (long)blockIdx.x * blockDim.x + threadIdx.x;
    int e = (int)(t >> 3);
    int g = (int)(t & 7);
    if (e >= E) return;
    int s = __builtin_nontemporal_load(ei + e);        // streamed once: NT
    int d = __builtin_nontemporal_load(ei + E + e);    // streamed once: NT
    float nrm = dinv[s] * dinv[d];
    const float4 v = *((const float4*)(h + (long)s * 32) + g);
    float* ap = agg + (long)d * 32 + g * 4;
    atomicAdd(ap + 0, v.x * nrm);
    atomicAdd(ap + 1, v.y * nrm);
    atomicAdd(ap + 2, v.z * nrm);
    atomicAdd(ap + 3, v.w * nrm);
}

// ---------------- self-loop + bias + ReLU (in place on agg) ----------------
__global__ void post_kernel(float* __restrict__ agg, const float* __restrict__ hlin,
                            const float* __restrict__ dinv, const float* __restrict__ bias,
                            int N, int F) {
    long t = (long)blockIdx.x * blockDim.x + threadIdx.x;
    if (t >= (long)N * F) return;
    int i = (int)(t / F);
    int f = (int)(t - (long)i * F);
    float di = dinv[i];
    float v = agg[t] + hlin[t] * di * di + bias[f];
    agg[t] = fmaxf(v, 0.0f);
}

// ---------------- pooling: sums[g][f] += h[i][f]; cnts[g] += 1 -------------
__global__ void pool_kernel(const float* __restrict__ h, const int* __restrict__ batch,
                            float* __restrict__ sums, float* __restrict__ cnts, int N) {
    long t = (long)blockIdx.x * blockDim.x + threadIdx.x;
    if (t >= (long)N * 32) return;
    int i = (int)(t >> 5);
    int f = (int)(t & 31);
    int g = __builtin_nontemporal_load(batch + i);     // streamed once: NT
    atomicAdd(&sums[(long)g * 32 + f], h[t]);
    if (f == 0) atomicAdd(&cnts[g], 1.0f);
}

// ---------------- head: sigmoid(mean_pool @ Wf + bf) -----------------------
__global__ void head_kernel(const float* __restrict__ sums, const float* __restrict__ cnts,
                            const float* __restrict__ Wf, const float* __restrict__ bf,
                            float* __restrict__ out, int G) {
    int g = blockIdx.x * blockDim.x + threadIdx.x;
    if (g >= G) return;
    float c = fmaxf(cnts[g], 1.0f);
    float inv = 1.0f / c;
    float acc = bf[0];
    #pragma unroll
    for (int f = 0; f < 32; ++f) {
        acc += sums[(long)g * 32 + f] * inv * Wf[f];
    }
    out[g] = 1.0f / (1.0f + __expf(-acc));
}

extern "C" void kernel_launch(void* const* d_in, const int* in_sizes, int n_in,
                              void* d_out, int out_size, void* d_ws, size_t ws_size,
                              hipStream_t stream) {
    const float* x   = (const float*)d_in[0];
    const int*   ei  = (const int*)d_in[1];     // [2, E] int32
    const int*   bat = (const int*)d_in[2];     // [N]
    const float* W1  = (const float*)d_in[3];
    const float* b1  = (const float*)d_in[4];
    const float* W2  = (const float*)d_in[5];
    const float* b2  = (const float*)d_in[6];
    const float* Wf  = (const float*)d_in[7];
    const float* bf  = (const float*)d_in[8];
    float* out = (float*)d_out;

    const int N = in_sizes[0] / 4;
    const int E = in_sizes[1] / 2;
    const int G = out_size;

    // Workspace layout: zeroed region first (one memset covers all atomics).
    float* ws = (float*)d_ws;
    size_t off = 0;
    float* deg  = ws + off; off += (size_t)N;        // zeroed
    float* agg1 = ws + off; off += (size_t)N * 16;   // zeroed
    float* agg2 = ws + off; off += (size_t)N * 32;   // zeroed
    float* sums = ws + off; off += (size_t)G * 32;   // zeroed
    float* cnts = ws + off; off += (size_t)G;        // zeroed
    size_t zero_floats = off;
    float* dinv = ws + off; off += (size_t)N;
    float* h1   = ws + off; off += (size_t)N * 16;
    float* h2   = ws + off; off += (size_t)N * 32;

    hipMemsetAsync(d_ws, 0, zero_floats * sizeof(float), stream);

    const int TB = 256;
    // 1) degree
    deg_kernel<<<(E + TB - 1) / TB, TB, 0, stream>>>(ei, deg, E);
    // 2) dinv
    dinv_kernel<<<(N + TB - 1) / TB, TB, 0, stream>>>(deg, dinv, N);
    // 3) layer-1 GEMM (WMMA): 1 tile per wave, 8 waves per block
    int ntiles = N / 16;
    l1_wmma_kernel<<<(ntiles + 7) / 8, TB, 0, stream>>>(x, W1, h1, ntiles);
    // 4) layer-1 edge scatter
    long t1 = (long)E * 4;
    edge16_kernel<<<(unsigned)((t1 + TB - 1) / TB), TB, 0, stream>>>(ei, dinv, h1, agg1, E);
    // 5) self-loop + bias + ReLU
    long p1 = (long)N * 16;
    post_kernel<<<(unsigned)((p1 + TB - 1) / TB), TB, 0, stream>>>(agg1, h1, dinv, b1, N, 16);
    // 6) layer-2 GEMM (WMMA)
    l2_wmma_kernel<<<(ntiles + 7) / 8, TB, 0, stream>>>(agg1, W2, h2, ntiles);
    // 7) layer-2 edge scatter
    long t2 = (long)E * 8;
    edge32_kernel<<<(unsigned)((t2 + TB - 1) / TB), TB, 0, stream>>>(ei, dinv, h2, agg2, E);
    // 8) self-loop + bias + ReLU
    long p2 = (long)N * 32;
    post_kernel<<<(unsigned)((p2 + TB - 1) / TB), TB, 0, stream>>>(agg2, h2, dinv, b2, N, 32);
    // 9) mean-pool scatter
    pool_kernel<<<(unsigned)((p2 + TB - 1) / TB), TB, 0, stream>>>(agg2, bat, sums, cnts, N);
    // 10) head
    head_kernel<<<(G + TB - 1) / TB, TB, 0, stream>>>(sums, cnts, Wf, bf, out, G);
}